// TFMamba3Block_16406775071291
// MI455X (gfx1250) — compile-verified
//
#include <hip/hip_runtime.h>
#include <math.h>

// ---------------------------------------------------------------------------
// Static problem configuration (matches reference)
// ---------------------------------------------------------------------------
#define DMODEL  64
#define DINNER  128
#define NHEADS  8
#define DSTATE  16
#define RANK    4
#define KCONV   4
#define DPROJX  520      // 2*DINNER + 2*NHEADS*DSTATE + NHEADS
#define BDIM    2
#define TDIM    192
#define FDIM    96
#define NTOK    36864    // BDIM*TDIM*FDIM (same token count for both axes)
#define WPADC   640      // in_w padded to 5 blocks x 128 cols
#define SCH     16       // scan chunk (timesteps staged in LDS)

typedef float v2f __attribute__((ext_vector_type(2)));
typedef float v8f __attribute__((ext_vector_type(8)));

__device__ __forceinline__ float silu_f(float x) { return x / (1.f + expf(-x)); }

// ---------------------------------------------------------------------------
// Relayout kernels
// ---------------------------------------------------------------------------
// x[B, C, T, F] -> X[(b*F+f), t, c]   (time-axis sequences)
__global__ void k_relayout_in(const float* __restrict__ x, float* __restrict__ X) {
    int idx = blockIdx.x * 256 + threadIdx.x;
    if (idx >= NTOK * DMODEL) return;
    int c = idx & 63;
    int r = idx >> 6;              // (b*F+f)*T + t
    int t = r % TDIM;
    int bf = r / TDIM;
    int f = bf % FDIM;
    int b = bf / FDIM;
    X[idx] = x[(((size_t)(b * DMODEL + c)) * TDIM + t) * FDIM + f];
}

// X1[(b*F+f), t, c] -> X2[(b*T+t), f, c]   (freq-axis sequences)
__global__ void k_relayout_mid(const float* __restrict__ X1, float* __restrict__ X2) {
    int idx = blockIdx.x * 256 + threadIdx.x;
    if (idx >= NTOK * DMODEL) return;
    int c = idx & 63;
    int r = idx >> 6;              // (b*T+t)*F + f
    int f = r % FDIM;
    int bt = r / FDIM;
    int t = bt % TDIM;
    int b = bt / TDIM;
    X2[idx] = X1[(((size_t)(b * FDIM + f)) * TDIM + t) * DMODEL + c];
}

// X[(b*T+t), f, c] -> out[B, C, T, F]
__global__ void k_relayout_out(const float* __restrict__ X, float* __restrict__ out) {
    int idx = blockIdx.x * 256 + threadIdx.x;
    if (idx >= NTOK * DMODEL) return;
    int f = idx % FDIM;
    int r = idx / FDIM;            // (b*64+c)*T + t
    int t = r % TDIM;
    int q = r / TDIM;
    int c = q & 63;
    int b = q >> 6;
    out[idx] = X[(((size_t)(b * TDIM + t)) * FDIM + f) * DMODEL + c];
}

// ---------------------------------------------------------------------------
// Zero-pad in_w [64 x 520] -> WPAD [64 x 640]
// ---------------------------------------------------------------------------
__global__ void k_pad_w(const float* __restrict__ W, float* __restrict__ WP) {
    int idx = blockIdx.x * 256 + threadIdx.x;
    if (idx >= DMODEL * WPADC) return;
    int c = idx % WPADC, k = idx / WPADC;
    WP[idx] = (c < DPROJX) ? W[k * DPROJX + c] : 0.f;
}

// ---------------------------------------------------------------------------
// LayerNorm over last dim (64), with optional per-sequence reversal of source
// Block = 256 threads = 4 tokens x 64 channels.
// ---------------------------------------------------------------------------
__global__ void k_layernorm(const float* __restrict__ X, float* __restrict__ O,
                            const float* __restrict__ gam, const float* __restrict__ bet,
                            int L, int rev) {
    __shared__ float red[256];
    int tid = threadIdx.x;
    int seg = tid >> 6, c = tid & 63;
    int tok = blockIdx.x * 4 + seg;
    int n = tok / L, t = tok % L;
    int ts = rev ? (L - 1 - t) : t;
    float v = X[((size_t)(n * L + ts)) * DMODEL + c];
    red[tid] = v; __syncthreads();
    for (int s = 32; s > 0; s >>= 1) { if (c < s) red[tid] += red[tid + s]; __syncthreads(); }
    float mean = red[seg * 64] * (1.f / 64.f);
    __syncthreads();
    float d = v - mean;
    red[tid] = d * d; __syncthreads();
    for (int s = 32; s > 0; s >>= 1) { if (c < s) red[tid] += red[tid + s]; __syncthreads(); }
    float var = red[seg * 64] * (1.f / 64.f);
    O[(size_t)tok * DMODEL + c] = d * rsqrtf(var + 1e-5f) * gam[c] + bet[c];
}

// ---------------------------------------------------------------------------
// fp32 WMMA GEMM:  Y = A[rows,K] @ W[K, cols] (+bias)(+resid), LDS-staged.
// Block = 8 waves arranged as MT x (8/MT) tiles of 16x16 output.
//  - A tile (MT*16 x K) and W panel (K x NT*16) staged in LDS via float4
//  - inner loop: pure ds_load + v_wmma_f32_16x16x4_f32, two acc chains
//  - W must be padded so blockIdx.y*NT*16 + NT*16 <= Wld (guard-free loads)
//  - store guards col < N; optional per-sequence row reversal (backward dir)
// ---------------------------------------------------------------------------
__global__ void k_gemm_f32(const float* __restrict__ A, int K,
                           const float* __restrict__ W, int Wld,
                           float* __restrict__ Y, int ldy, int N,
                           const float* __restrict__ bias,
                           const float* __restrict__ resid, int ldr,
                           int L, int rev, int MT) {
    const int ntsh = (MT == 1) ? 3 : 2;     // log2(NT)
    const int NT   = 1 << ntsh;
    const int NTC  = NT << 4;               // cols per block: 128 or 64
    extern __shared__ float lds[];
    float* As = lds;                        // [MT*16][K]
    float* Ws = lds + MT * 16 * K;          // [K][NTC]

    int tid = threadIdx.x;
    int mbase = blockIdx.x * MT * 16;
    int cbase = blockIdx.y * NTC;

    // Cooperative stage: A rows are contiguous
    {
        const float4* A4 = (const float4*)(A + (size_t)mbase * K);
        float4* As4 = (float4*)As;
        int an4 = (MT * 16 * K) >> 2;
        for (int i = tid; i < an4; i += 256) As4[i] = A4[i];
    }
    // Cooperative stage: W panel (row stride Wld)
    {
        const float4* W4 = (const float4*)(W + cbase);
        float4* Ws4 = (float4*)Ws;
        int ncsh = ntsh + 2;                // log2(NTC/4)
        int wn4 = K << ncsh;
        int wld4 = Wld >> 2;
        for (int i = tid; i < wn4; i += 256) {
            int kk = i >> ncsh, c4 = i & ((1 << ncsh) - 1);
            Ws4[i] = W4[(size_t)kk * wld4 + c4];
        }
    }
    __syncthreads();

    int wave = tid >> 5, lane = tid & 31;
    int mw = wave >> ntsh, nw = wave & (NT - 1);
    int g = lane >> 4, ml = lane & 15;

    int aoff = (mw * 16 + ml) * K + 2 * g;   // A frag: M=ml(+16*mw), K=k0+2g+{0,1}
    int woff = (2 * g) * NTC + nw * 16 + ml; // B frag: K=k0+2g+v, N=ml

    v8f acc0 = {0.f, 0.f, 0.f, 0.f, 0.f, 0.f, 0.f, 0.f};
    v8f acc1 = {0.f, 0.f, 0.f, 0.f, 0.f, 0.f, 0.f, 0.f};
    for (int k0 = 0; k0 < K; k0 += 8) {
        v2f a0 = *(const v2f*)(As + aoff);
        v2f b0; b0.x = Ws[woff];            b0.y = Ws[woff + NTC];
        acc0 = __builtin_amdgcn_wmma_f32_16x16x4_f32(false, a0, false, b0,
                                                     (short)0, acc0, false, false);
        v2f a1 = *(const v2f*)(As + aoff + 4);
        v2f b1; b1.x = Ws[woff + 4 * NTC];  b1.y = Ws[woff + 5 * NTC];
        acc1 = __builtin_amdgcn_wmma_f32_16x16x4_f32(false, a1, false, b1,
                                                     (short)0, acc1, false, false);
        aoff += 8;
        woff += 8 * NTC;
    }

    int col = cbase + nw * 16 + ml;
    if (col >= N) return;
    float bv = bias ? bias[col] : 0.f;
    int m0 = mbase + mw * 16;
#pragma unroll
    for (int r = 0; r < 8; ++r) {
        int row = m0 + r + 8 * g;           // D frag: M = r + 8g
        int n = row / L, t = row % L;
        int rowo = rev ? (n * L + (L - 1 - t)) : row;
        float v = acc0[r] + acc1[r] + bv;
        if (resid) v += resid[(size_t)rowo * ldr + col];
        Y[(size_t)rowo * ldy + col] = v;
    }
}

// ---------------------------------------------------------------------------
// Causal depthwise conv (K=4) + SiLU on the xs slice of PROJ
// ---------------------------------------------------------------------------
__global__ void k_conv_silu(const float* __restrict__ PROJ,
                            const float* __restrict__ cw, const float* __restrict__ cb,
                            float* __restrict__ XSC, int L) {
    int idx = blockIdx.x * 256 + threadIdx.x;
    if (idx >= NTOK * DINNER) return;
    int c = idx & 127;
    int tok = idx >> 7;
    int n = tok / L, t = tok % L;
    float acc = cb[c];
#pragma unroll
    for (int k = 0; k < KCONV; ++k) {
        int tp = t - (KCONV - 1) + k;
        if (tp >= 0) acc += cw[k * DINNER + c] * PROJ[(size_t)(n * L + tp) * DPROJX + c];
    }
    XSC[idx] = silu_f(acc);
}

// softplus(dt + dt_bias)
__global__ void k_dt(const float* __restrict__ PROJ, const float* __restrict__ dtb,
                     float* __restrict__ DTB) {
    int idx = blockIdx.x * 256 + threadIdx.x;
    if (idx >= NTOK * NHEADS) return;
    int h = idx & 7;
    int tok = idx >> 3;
    float x = PROJ[(size_t)tok * DPROJX + 2 * DINNER + 2 * NHEADS * DSTATE + h] + dtb[h];
    DTB[idx] = (x > 20.f) ? x : log1pf(expf(x));
}

// u[tok,h,r] = sum_d xh[tok,h,d] * P[h,d,r]   (rank-R MIMO in-mixer)
__global__ void k_u(const float* __restrict__ XSC, const float* __restrict__ P,
                    float* __restrict__ UB) {
    int idx = blockIdx.x * 256 + threadIdx.x;
    if (idx >= NTOK * NHEADS * RANK) return;
    int r = idx & 3;
    int h = (idx >> 2) & 7;
    int tok = idx >> 5;
    float acc = 0.f;
#pragma unroll
    for (int d = 0; d < DSTATE; ++d)
        acc += XSC[(size_t)tok * DINNER + h * DSTATE + d] * P[(h * DSTATE + d) * RANK + r];
    UB[idx] = acc;
}

// ---------------------------------------------------------------------------
// Bilinear-discretized complex selective scan + C-readout + Q out-mixer.
// One block per sequence; wave w = head w; lane holds (s = lane>>1, two r's).
// SCH timesteps of B/C/dt/u are staged in LDS per barrier so the recurrence
// critical path only pays one global round-trip per chunk.
// ---------------------------------------------------------------------------
__global__ void k_scan(const float* __restrict__ PROJ, const float* __restrict__ DTB,
                       const float* __restrict__ UB,
                       const float* __restrict__ A_log, const float* __restrict__ A_imag,
                       const float* __restrict__ Q, float* __restrict__ YS, int L) {
    __shared__ float sB[SCH * 128];
    __shared__ float sC[SCH * 128];
    __shared__ float sdt[SCH * 8];
    __shared__ float su[SCH * 32];

    int n = blockIdx.x;
    int tid = threadIdx.x;
    int h = tid >> 5, lane = tid & 31;
    int s = lane >> 1, p = lane & 1;
    int d = lane & 15;

    float Ar = -expf(A_log[h * DSTATE + s]);
    float Ai = A_imag[h * DSTATE + s];
    float q0 = Q[(h * RANK + 0) * DSTATE + d];
    float q1 = Q[(h * RANK + 1) * DSTATE + d];
    float q2 = Q[(h * RANK + 2) * DSTATE + d];
    float q3 = Q[(h * RANK + 3) * DSTATE + d];

    float S0r = 0.f, S0i = 0.f, S1r = 0.f, S1i = 0.f;   // states for r = 2p, 2p+1
    float Bp = 0.f, u0p = 0.f, u1p = 0.f;               // t-1 values (zero pad at t=0)

    for (int c0 = 0; c0 < L; c0 += SCH) {
        // ---- stage chunk into LDS ----
        for (int i = tid; i < SCH * 128; i += 256) {
            int tt = i >> 7, c = i & 127;
            size_t base = (size_t)(n * L + c0 + tt) * DPROJX + 2 * DINNER;
            sB[i] = PROJ[base + c];
            sC[i] = PROJ[base + NHEADS * DSTATE + c];
            if (c0 + SCH < L)   // pull next chunk toward L2/L0 (global_prefetch_b8)
                __builtin_prefetch(&PROJ[base + (size_t)SCH * DPROJX + c], 0, 1);
        }
        for (int i = tid; i < SCH * 32; i += 256) {
            int tt = i >> 5, c = i & 31;
            su[i] = UB[(size_t)(n * L + c0 + tt) * (NHEADS * RANK) + c];
        }
        if (tid < SCH * 8) {
            int tt = tid >> 3, hh = tid & 7;
            sdt[tid] = DTB[(size_t)(n * L + c0 + tt) * NHEADS + hh];
        }
        __syncthreads();

        // ---- SCH sequential recurrence steps out of LDS ----
        for (int tt = 0; tt < SCH; ++tt) {
            float dt = sdt[tt * 8 + h];
            float Bt = sB[tt * 128 + h * DSTATE + s];
            float Ct = sC[tt * 128 + h * DSTATE + s];
            float u0 = su[tt * 32 + h * RANK + 2 * p];
            float u1 = su[tt * 32 + h * RANK + 2 * p + 1];

            // dA = dt*A ; denom = 1 - dA/2 ; alpha = (1+dA/2)/denom ; coef = dt/(2*denom)
            float hdr = 0.5f * dt * Ar, hdi = 0.5f * dt * Ai;
            float denr = 1.f - hdr, deni = -hdi;
            float inv = 1.f / (denr * denr + deni * deni);
            float numr = 1.f + hdr, numi = hdi;
            float alr = (numr * denr + numi * deni) * inv;
            float ali = (numi * denr - numr * deni) * inv;
            float cr = 0.5f * dt * denr * inv;
            float ci = -0.5f * dt * deni * inv;

            float inj0 = Bt * u0 + Bp * u0p;    // trapezoidal drive (real)
            float inj1 = Bt * u1 + Bp * u1p;

            float t0r = alr * S0r - ali * S0i + cr * inj0;
            float t0i = alr * S0i + ali * S0r + ci * inj0;
            float t1r = alr * S1r - ali * S1i + cr * inj1;
            float t1i = alr * S1i + ali * S1r + ci * inj1;
            S0r = t0r; S0i = t0i; S1r = t1r; S1i = t1i;

            // yr[h,r] = Re( sum_s C * S ) — xor-tree over s (lane bits 1..4)
            float pa = Ct * S0r;
            float pb = Ct * S1r;
#pragma unroll
            for (int m = 2; m <= 16; m <<= 1) {
                pa += __shfl_xor(pa, m, 32);
                pb += __shfl_xor(pb, m, 32);
            }
            float yr0 = __shfl(pa, 0, 32);   // r=0 (p=0 lane)
            float yr1 = __shfl(pb, 0, 32);   // r=1
            float yr2 = __shfl(pa, 1, 32);   // r=2 (p=1 lane)
            float yr3 = __shfl(pb, 1, 32);   // r=3
            if (lane < 16)
                YS[(size_t)(n * L + c0 + tt) * DINNER + h * DSTATE + d] =
                    yr0 * q0 + yr1 * q1 + yr2 * q2 + yr3 * q3;

            Bp = Bt; u0p = u0; u1p = u1;
        }
        __syncthreads();
    }
}

// y = (y + D*xs) * silu(z)   (in-place on YS)
__global__ void k_gate(float* __restrict__ YS, const float* __restrict__ XSC,
                       const float* __restrict__ PROJ, const float* __restrict__ Dsk) {
    int idx = blockIdx.x * 256 + threadIdx.x;
    if (idx >= NTOK * DINNER) return;
    int c = idx & 127;
    int tok = idx >> 7;
    float z = PROJ[(size_t)tok * DPROJX + DINNER + c];
    YS[idx] = (YS[idx] + Dsk[c] * XSC[idx]) * silu_f(z);
}

// ---------------------------------------------------------------------------
// Host-side parameter tables (two possible flatten orders, auto-detected)
// ---------------------------------------------------------------------------
struct MambaP {
    const float *in_w, *conv_w, *conv_b, *dt_bias, *A_log, *A_imag, *D, *P, *Q, *out_w, *out_b;
};
struct BiP {
    const float *ln_f_g, *ln_f_b, *ln_b_g, *ln_b_b;
    MambaP mf, mb;
};

static MambaP mamba_ins(void* const* d, int b) {
    MambaP m;
    m.in_w = (const float*)d[b + 0];   m.conv_w = (const float*)d[b + 1];
    m.conv_b = (const float*)d[b + 2]; m.dt_bias = (const float*)d[b + 3];
    m.A_log = (const float*)d[b + 4];  m.A_imag = (const float*)d[b + 5];
    m.D = (const float*)d[b + 6];      m.P = (const float*)d[b + 7];
    m.Q = (const float*)d[b + 8];      m.out_w = (const float*)d[b + 9];
    m.out_b = (const float*)d[b + 10];
    return m;
}
static MambaP mamba_alpha(void* const* d, int b) {
    MambaP m;
    m.A_imag = (const float*)d[b + 0]; m.A_log = (const float*)d[b + 1];
    m.D = (const float*)d[b + 2];      m.P = (const float*)d[b + 3];
    m.Q = (const float*)d[b + 4];      m.conv_b = (const float*)d[b + 5];
    m.conv_w = (const float*)d[b + 6]; m.dt_bias = (const float*)d[b + 7];
    m.in_w = (const float*)d[b + 8];   m.out_b = (const float*)d[b + 9];
    m.out_w = (const float*)d[b + 10];
    return m;
}
static BiP bi_ins(void* const* d, int b) {
    BiP p;
    p.ln_f_g = (const float*)d[b + 0]; p.ln_f_b = (const float*)d[b + 1];
    p.ln_b_g = (const float*)d[b + 2]; p.ln_b_b = (const float*)d[b + 3];
    p.mf = mamba_ins(d, b + 4);
    p.mb = mamba_ins(d, b + 15);
    return p;
}
static BiP bi_alpha(void* const* d, int b) {
    BiP p;
    p.ln_b_b = (const float*)d[b + 0]; p.ln_b_g = (const float*)d[b + 1];
    p.ln_f_b = (const float*)d[b + 2]; p.ln_f_g = (const float*)d[b + 3];
    p.mb = mamba_alpha(d, b + 4);
    p.mf = mamba_alpha(d, b + 15);
    return p;
}

// ---------------------------------------------------------------------------
// One bidirectional mamba block over one axis:  XN = bimamba(XC) @ lin + b + XC
// ---------------------------------------------------------------------------
static void run_bimamba(int BN, int L, const BiP& bp,
                        const float* lin_w, const float* lin_b,
                        const float* XC, float* XN,
                        float* LNX, float* PROJ, float* XSC, float* DTB,
                        float* UB, float* YS, float* Y2, float* WPAD,
                        hipStream_t stream) {
    const size_t lds_in  = (16 * DMODEL + DMODEL * 128) * sizeof(float);   // 36 KB
    const size_t lds_out = (32 * DINNER + DINNER * 64) * sizeof(float);    // 48 KB
    for (int dir = 0; dir < 2; ++dir) {
        const MambaP& mp = dir ? bp.mb : bp.mf;
        const float* g = dir ? bp.ln_b_g : bp.ln_f_g;
        const float* be = dir ? bp.ln_b_b : bp.ln_f_b;

        k_layernorm<<<NTOK / 4, 256, 0, stream>>>(XC, LNX, g, be, L, dir);
        // in-proj: [NTOK,64] x [64,520] via zero-padded [64,640] weights
        k_pad_w<<<(DMODEL * WPADC) / 256, 256, 0, stream>>>(mp.in_w, WPAD);
        k_gemm_f32<<<dim3(NTOK / 16, WPADC / 128), 256, lds_in, stream>>>(
            LNX, DMODEL, WPAD, WPADC, PROJ, DPROJX, DPROJX, nullptr, nullptr, 0,
            L, 0, /*MT=*/1);
        k_conv_silu<<<(NTOK * DINNER) / 256, 256, 0, stream>>>(PROJ, mp.conv_w, mp.conv_b, XSC, L);
        k_dt<<<(NTOK * NHEADS) / 256, 256, 0, stream>>>(PROJ, mp.dt_bias, DTB);
        k_u<<<(NTOK * NHEADS * RANK) / 256, 256, 0, stream>>>(XSC, mp.P, UB);
        k_scan<<<BN, 256, 0, stream>>>(PROJ, DTB, UB, mp.A_log, mp.A_imag, mp.Q, YS, L);
        k_gate<<<(NTOK * DINNER) / 256, 256, 0, stream>>>(YS, XSC, PROJ, mp.D);
        // out-proj: [NTOK,128] x [128,64] -> Y2 columns [dir*64, dir*64+64),
        // stored with per-sequence reversal for the backward direction.
        k_gemm_f32<<<dim3(NTOK / 32, 1), 256, lds_out, stream>>>(
            YS, DINNER, mp.out_w, DMODEL, Y2 + dir * DMODEL, DINNER, DMODEL,
            mp.out_b, nullptr, 0, L, dir, /*MT=*/2);
    }
    // axis linear + residual:  XN = Y2 @ lin_w + lin_b + XC
    k_gemm_f32<<<dim3(NTOK / 32, 1), 256, lds_out, stream>>>(
        Y2, DINNER, lin_w, DMODEL, XN, DMODEL, DMODEL, lin_b, XC, DMODEL,
        L, 0, /*MT=*/2);
}

// ---------------------------------------------------------------------------
extern "C" void kernel_launch(void* const* d_in, const int* in_sizes, int n_in,
                              void* d_out, int out_size, void* d_ws, size_t ws_size,
                              hipStream_t stream) {
    (void)n_in; (void)out_size; (void)ws_size;

    // Detect flatten order: insertion order puts x (2359296 elems) first;
    // jax-style alphabetical flatten puts it last.
    const bool insertion = (in_sizes[0] == NTOK * DMODEL);

    const float* x;
    BiP timeP, freqP;
    const float *tlin_w, *tlin_b, *flin_w, *flin_b;
    if (insertion) {
        x = (const float*)d_in[0];
        timeP = bi_ins(d_in, 1);
        freqP = bi_ins(d_in, 27);
        tlin_w = (const float*)d_in[53]; tlin_b = (const float*)d_in[54];
        flin_w = (const float*)d_in[55]; flin_b = (const float*)d_in[56];
    } else {
        flin_b = (const float*)d_in[0];  flin_w = (const float*)d_in[1];
        freqP = bi_alpha(d_in, 2);
        timeP = bi_alpha(d_in, 28);
        tlin_b = (const float*)d_in[54]; tlin_w = (const float*)d_in[55];
        x = (const float*)d_in[56];
    }

    // Workspace carve-up (floats)
    float* ws = (float*)d_ws;
    size_t o = 0;
    float* XC   = ws + o; o += (size_t)NTOK * DMODEL;   // current axis activations
    float* XN   = ws + o; o += (size_t)NTOK * DMODEL;   // next axis activations
    float* LNX  = ws + o; o += (size_t)NTOK * DMODEL;   // layernormed (reversed) input
    float* PROJ = ws + o; o += (size_t)NTOK * DPROJX;   // in-proj output [xs|z|B|C|dt]
    float* XSC  = ws + o; o += (size_t)NTOK * DINNER;   // conv+silu(xs)
    float* DTB  = ws + o; o += (size_t)NTOK * NHEADS;   // softplus(dt)
    float* UB   = ws + o; o += (size_t)NTOK * NHEADS * RANK; // rank-R input u
    float* YS   = ws + o; o += (size_t)NTOK * DINNER;   // scan output -> gated in place
    float* Y2   = ws + o; o += (size_t)NTOK * DINNER;   // bimamba concat [yf|yb]
    float* WPAD = ws + o; o += (size_t)DMODEL * WPADC;  // zero-padded in_w

    const int NE = NTOK * DMODEL;

    // ---- time axis ----
    k_relayout_in<<<NE / 256, 256, 0, stream>>>(x, XC);
    run_bimamba(BDIM * FDIM, TDIM, timeP, tlin_w, tlin_b, XC, XN,
                LNX, PROJ, XSC, DTB, UB, YS, Y2, WPAD, stream);

    // ---- freq axis ----
    k_relayout_mid<<<NE / 256, 256, 0, stream>>>(XN, XC);
    run_bimamba(BDIM * TDIM, FDIM, freqP, flin_w, flin_b, XC, XN,
                LNX, PROJ, XSC, DTB, UB, YS, Y2, WPAD, stream);

    // ---- back to [B, C, T, F] ----
    k_relayout_out<<<NE / 256, 256, 0, stream>>>(XN, (float*)d_out);
}